// FlashMoE_35527969472974
// MI455X (gfx1250) — compile-verified
//
#include <hip/hip_runtime.h>
#include <stdint.h>

#define B_  8
#define S_  1024
#define D_  768
#define E_  24
#define DH_ 128
#define DF_ 384
#define N_  6144           // S_ * H_
#define K_  1024           // tokens per expert
#define BS_ 8192           // B_ * S_

typedef float v2f __attribute__((ext_vector_type(2)));
typedef float v8f __attribute__((ext_vector_type(8)));

// ---------------------------------------------------------------------------
// CDNA5 fp32 WMMA: D(16x16,f32) = A(16x4,f32) * B(4x16,f32) + C
// A layout : lane l(0-15) -> row; VGPR r, half h -> K = 2h+r
// B layout : lane l -> col;       VGPR r, half h -> K = 2h+r
// C/D      : VGPR r, half h -> row = r + 8h, lane l -> col
// ---------------------------------------------------------------------------
__device__ __forceinline__ v8f wmma4(v2f a, v2f b, v8f c) {
  return __builtin_amdgcn_wmma_f32_16x16x4_f32(
      /*neg_a=*/false, a, /*neg_b=*/false, b,
      /*c_mod=*/(short)0, c, /*reuse_a=*/false, /*reuse_b=*/false);
}

__device__ __forceinline__ v2f load_a_frag(const float* p, int lda) {
  const int lane = threadIdx.x & 31;
  const float* q = p + (size_t)(lane & 15) * lda + ((lane >> 4) << 1);
  v2f a; a.x = q[0]; a.y = q[1];
  return a;
}

__device__ __forceinline__ v2f load_b_frag(const float* p, int ldb) {
  const int lane = threadIdx.x & 31;
  const float* q = p + (size_t)((lane >> 4) << 1) * ldb + (lane & 15);
  v2f b; b.x = q[0]; b.y = q[ldb];
  return b;
}

// ---------------------------------------------------------------------------
// gfx1250 async global->LDS copy (ASYNCcnt-tracked, no VGPR round trip).
// Generic __shared__ pointers carry the LDS offset in their low 32 bits.
// ---------------------------------------------------------------------------
__device__ __forceinline__ uint32_t lds_off32(const void* p) {
  return (uint32_t)(uintptr_t)p;
}
__device__ __forceinline__ void async_b128_to_lds(uint32_t lds, const void* g) {
  asm volatile("global_load_async_to_lds_b128 %0, %1, off"
               :: "v"(lds), "v"(g) : "memory");
}
__device__ __forceinline__ void wait_async0() {
  asm volatile("s_wait_asynccnt 0" ::: "memory");
}

// ---------------------------------------------------------------------------
// Dense fp32 GEMM: C[M,Nc] = A[M,K] @ B[K,Nc]   (row-major)
// grid = (M/16, Nc/128), block = 256 (8 wave32) -> one 16x16 tile per wave.
// The 16xK A panel is shared by all 8 waves: async-stage 16x64 chunks in LDS
// (stride 68 floats: 16B-aligned rows, bank-conflict-free fragment reads).
// ---------------------------------------------------------------------------
__global__ __launch_bounds__(256) void gemm_kernel(
    const float* __restrict__ A, const float* __restrict__ Bm,
    float* __restrict__ C, int M, int K, int Nc) {
  __shared__ float as[16][68];

  const int tid  = threadIdx.x;
  const int wave = tid >> 5;
  const int row0 = blockIdx.x * 16;
  const int col0 = (blockIdx.y * 8 + wave) * 16;

  // cooperative fill: one b128 (4 floats) per thread per chunk
  const int arow = tid >> 4;          // 0..15
  const int acol = (tid & 15) << 2;   // 0,4,...,60
  const uint32_t a_lds   = lds_off32(&as[arow][acol]);
  const float*   a_gbase = A + (size_t)(row0 + arow) * K + acol;

  v8f acc = {};
  for (int kc = 0; kc < K; kc += 64) {
    __syncthreads();                       // prior chunk fully consumed
    async_b128_to_lds(a_lds, a_gbase + kc);
    if (kc + 64 < K)                       // wave-uniform L2 prefetch of next B
      __builtin_prefetch(Bm + (size_t)(kc + 64) * Nc + col0, 0, 0);
    wait_async0();
    __syncthreads();                       // chunk visible to all waves
#pragma unroll
    for (int k = 0; k < 64; k += 4) {
      v2f a = load_a_frag(&as[0][k], 68);
      v2f b = load_b_frag(Bm + (size_t)(kc + k) * Nc + col0, Nc);
      acc = wmma4(a, b, acc);
    }
  }

  const int lane = threadIdx.x & 31;
  const int half = lane >> 4, l = lane & 15;
  float* Cp = C + (size_t)row0 * Nc + col0;
#pragma unroll
  for (int r = 0; r < 8; ++r)
    Cp[(size_t)(r + 8 * half) * Nc + l] = acc[r];
}

// ---------------------------------------------------------------------------
// gates[b,e,n] = softmax_e( y[b,n,:] . choice[e,:] )   stored [B,E,N]
// ---------------------------------------------------------------------------
__global__ __launch_bounds__(256) void gates_kernel(
    const float* __restrict__ y, const float* __restrict__ choice,
    float* __restrict__ gates) {
  const int t = blockIdx.x * 256 + threadIdx.x;   // token in [0, B_*N_)
  const int b = t / N_;
  const int n = t - b * N_;
  const float* row = y + (size_t)t * DH_;

  float z[E_];
  float m = -3.402823e38f;
#pragma unroll 1
  for (int e = 0; e < E_; ++e) {
    const float* ce = choice + e * DH_;
    float s = 0.f;
    for (int d = 0; d < DH_; ++d) s = fmaf(row[d], ce[d], s);
    z[e] = s;
    m = fmaxf(m, s);
  }
  float sum = 0.f;
#pragma unroll
  for (int e = 0; e < E_; ++e) { z[e] = __expf(z[e] - m); sum += z[e]; }
  const float inv = 1.f / sum;
#pragma unroll
  for (int e = 0; e < E_; ++e)
    gates[((size_t)b * E_ + e) * N_ + n] = z[e] * inv;
}

// ---------------------------------------------------------------------------
// Exact top-K by rank counting; reproduces jax.lax.top_k order + tie-breaks.
// ---------------------------------------------------------------------------
__global__ __launch_bounds__(256) void topk_kernel(
    const float* __restrict__ gates, int* __restrict__ I,
    float* __restrict__ G) {
  __shared__ float g[N_];
  const int be = blockIdx.x;
  const float* grow = gates + (size_t)be * N_;
  for (int i = threadIdx.x; i < N_; i += 256) g[i] = grow[i];
  __syncthreads();

  for (int i = threadIdx.x; i < N_; i += 256) {
    const float gi = g[i];
    int rank = 0;
    for (int j = 0; j < N_; ++j) {
      const float gj = g[j];
      rank += (gj > gi) || (gj == gi && j < i);
    }
    if (rank < K_) {
      I[(size_t)be * K_ + rank] = i;
      G[(size_t)be * K_ + rank] = gi;
    }
  }
}

// ---------------------------------------------------------------------------
// Expert FFN + weighted scatter-add.
// block = 16 tokens of one (b,e); 256 threads = 8 waves.
//   xs : gathered inputs 16 x 128, async-loaded (stride 132: rows 16B-aligned,
//        fragment reads conflict-free since banks = (4l+k) mod 64)
//   hs : silu(xs @ w1) 16 x 384 (stride 385, conflict-free)
//   out = hs @ w2 -> gate-weighted f32 atomic scatter into xout
// ---------------------------------------------------------------------------
__global__ __launch_bounds__(256) void ffn_kernel(
    const float* __restrict__ y, const float* __restrict__ w1,
    const float* __restrict__ w2, const int* __restrict__ I,
    const float* __restrict__ G, float* __restrict__ xout) {
  __shared__ float xs[16][DH_ + 4];   // 16 x 132
  __shared__ float hs[16][DF_ + 1];   // 16 x 385
  __shared__ int   tok[16];
  __shared__ float gv[16];

  const int rblk = blockIdx.x & (K_ / 16 - 1);   // K_/16 == 64
  const int be   = blockIdx.x >> 6;
  const int e    = be % E_;
  const int b    = be / E_;
  const int r0   = rblk * 16;

  const int tid = threadIdx.x;
  if (tid < 16) {
    tok[tid] = I[(size_t)be * K_ + r0 + tid];
    gv[tid]  = G[(size_t)be * K_ + r0 + tid];
  }
  __syncthreads();

  // async-gather 16 token rows (16 x 128 f32): 512 x b128, 2 per thread
  {
    const uint32_t xs_base = lds_off32(&xs[0][0]);
#pragma unroll
    for (int i = 0; i < 2; ++i) {
      const int q   = tid + i * 256;     // 0..511
      const int r   = q >> 5;            // row 0..15
      const int c   = (q & 31) << 2;     // col 0,4,...,124
      const float* g = y + ((size_t)b * N_ + tok[r]) * DH_ + c;
      async_b128_to_lds(xs_base + (uint32_t)(r * (DH_ + 4) + c) * 4u, g);
    }
    wait_async0();
  }
  __syncthreads();

  const int wave = tid >> 5;
  const int lane = tid & 31;
  const int half = lane >> 4, l = lane & 15;
  const float* w1e = w1 + (size_t)e * DH_ * DF_;   // [DH_, DF_]
  const float* w2e = w2 + (size_t)e * DF_ * DH_;   // [DF_, DH_]

  // h = silu(xs @ w1e) : 16 x 384 -> 24 col-tiles, 3 per wave
  for (int ct = wave; ct < DF_ / 16; ct += 8) {
    const int col0 = ct * 16;
    v8f acc = {};
#pragma unroll
    for (int k = 0; k < DH_; k += 4) {
      v2f a  = load_a_frag(&xs[0][k], DH_ + 4);
      v2f bf = load_b_frag(w1e + (size_t)k * DF_ + col0, DF_);
      acc = wmma4(a, bf, acc);
    }
#pragma unroll
    for (int r = 0; r < 8; ++r) {
      const float v = acc[r];
      hs[r + 8 * half][col0 + l] = v / (1.f + __expf(-v));   // silu
    }
  }
  __syncthreads();

  // out = hs @ w2e : 16 x 128 -> 8 col-tiles, 1 per wave; gated scatter
  {
    const int col0 = wave * 16;
    v8f acc = {};
    for (int k = 0; k < DF_; k += 4) {
      v2f a  = load_a_frag(&hs[0][k], DF_ + 1);
      v2f bf = load_b_frag(w2e + (size_t)k * DH_ + col0, DH_);
      acc = wmma4(a, bf, acc);
    }
#pragma unroll
    for (int r = 0; r < 8; ++r) {
      const int row = r + 8 * half;
      atomicAdd(&xout[((size_t)b * N_ + tok[row]) * DH_ + col0 + l],
                gv[row] * acc[r]);
    }
  }
}

// ---------------------------------------------------------------------------
extern "C" void kernel_launch(void* const* d_in, const int* in_sizes, int n_in,
                              void* d_out, int out_size, void* d_ws,
                              size_t ws_size, hipStream_t stream) {
  const float* x      = (const float*)d_in[0];   // [B,S,D]
  const float* choice = (const float*)d_in[1];   // [E,DH]
  const float* w1     = (const float*)d_in[2];   // [E,DH,DF]
  const float* w2     = (const float*)d_in[3];   // [E,DF,DH]
  const float* head   = (const float*)d_in[4];   // [D,D]
  const float* merge  = (const float*)d_in[5];   // [D,D]
  float* out = (float*)d_out;                    // [B,S,D]

  char* ws = (char*)d_ws;
  float* y     = (float*)ws; ws += (size_t)BS_ * D_ * sizeof(float);
  float* gates = (float*)ws; ws += (size_t)B_ * E_ * N_ * sizeof(float);
  int*   I     = (int*)ws;   ws += (size_t)B_ * E_ * K_ * sizeof(int);
  float* G     = (float*)ws; ws += (size_t)B_ * E_ * K_ * sizeof(float);
  float* xout  = (float*)ws; ws += (size_t)BS_ * D_ * sizeof(float);
  (void)in_sizes; (void)n_in; (void)out_size; (void)ws_size;

  // 1. y = x @ head   (flat [8192,768] @ [768,768])
  gemm_kernel<<<dim3(BS_ / 16, D_ / 128), 256, 0, stream>>>(
      x, head, y, BS_, D_, D_);

  // 2. routing gates (softmax over experts), stored [B,E,N]
  gates_kernel<<<(B_ * N_) / 256, 256, 0, stream>>>(y, choice, gates);

  // 3. expert-choice top-K (exact, tie-break identical to jax.lax.top_k)
  topk_kernel<<<B_ * E_, 256, 0, stream>>>(gates, I, G);

  // 4. zero scatter target (async memset is graph-capture legal)
  hipMemsetAsync(xout, 0, (size_t)BS_ * D_ * sizeof(float), stream);

  // 5. per-expert FFN + gated scatter-add
  ffn_kernel<<<B_ * E_ * (K_ / 16), 256, 0, stream>>>(y, w1, w2, I, G, xout);

  // 6. out = xout @ merge
  gemm_kernel<<<dim3(BS_ / 16, D_ / 128), 256, 0, stream>>>(
      xout, merge, out, BS_, D_, D_);
}